// ConvSPE_57707180589131
// MI455X (gfx1250) — compile-verified
//
#include <hip/hip_runtime.h>
#include <stdint.h>

// ---------------- problem constants ----------------
#define RDIM   64
#define SDIM   2048
#define KTAP   201
#define KPAD   224        // 7 chunks * 32, zero-padded taps
#define CDIM   512
#define PADLEN 2852
#define NCHUNK 7
#define KC     32
#define SB     32         // s per block (and per thread)
#define CB     8          // channels per block
#define RBLK   32         // r per block
#define WIN    63         // SB + KC - 1 timesteps per chunk
#define TPAD   68         // padded t-stride (dwords), mult of 4, 2-way banks
#define CSTR   2184       // padded c-stride (dwords) = 32*68 + 8

typedef __attribute__((address_space(1))) int global_int;
typedef __attribute__((address_space(3))) int lds_int;

// ---------------- CDNA5 async global->LDS ----------------
__device__ __forceinline__ void async_load_b32(const float* g, float* l) {
#if defined(__gfx1250__) && __has_builtin(__builtin_amdgcn_global_load_async_to_lds_b32)
  __builtin_amdgcn_global_load_async_to_lds_b32(
      (global_int*)g, (lds_int*)l, 0, 0);
#else
  unsigned lo = (unsigned)(uintptr_t)(lds_int*)l;
  unsigned long long ga = (unsigned long long)(uintptr_t)g;
  asm volatile("global_load_async_to_lds_b32 %0, %1, off"
               :: "v"(lo), "v"(ga) : "memory");
#endif
}

__device__ __forceinline__ void wait_async0() {
#if defined(__gfx1250__) && __has_builtin(__builtin_amdgcn_s_wait_asynccnt)
  __builtin_amdgcn_s_wait_asynccnt(0);
#else
  asm volatile("s_wait_asynccnt 0" ::: "memory");
#endif
}

// ---------------- kernel ----------------
__global__ __launch_bounds__(256) void ConvSPE_kernel(
    const float* __restrict__ z, const float* __restrict__ w_q,
    const float* __restrict__ w_k, float* __restrict__ outA,
    float* __restrict__ outB)
{
  // double-buffered z tile [buf][c][r][t] with padded strides, + padded weights
  __shared__ __align__(16) float zbuf[2][CB * CSTR];          // 2 * 69.9 KB
  __shared__ __align__(16) float wlds[2 * CB * KPAD];         // 14.3 KB

  const int tid = threadIdx.x;
  const int bx  = blockIdx.x;
  const int cb  = bx & 63;          // channel tile  (512/8  = 64)
  const int sb  = (bx >> 6) & 63;   // s tile        (2048/32 = 64)
  const int rb  = bx >> 12;         // r tile        (64/32   = 2)
  const int s0  = sb * SB;
  const int c0  = cb * CB;

  // ---- stage zero-padded weights into LDS: wlds[f][c][k]
  // filter 0 = w_k -> first reference output, filter 1 = w_q -> second
  for (int m = tid; m < 2 * CB * KPAD; m += 256) {
    const int f   = m / (CB * KPAD);
    const int rem = m - f * (CB * KPAD);
    const int cc  = rem / KPAD;
    const int k   = rem - cc * KPAD;
    const float* src = (f == 0) ? w_k : w_q;
    wlds[m] = (k < KTAP) ? src[(size_t)k * CDIM + c0 + cc] : 0.0f;
  }

  // ---- async-fill chunk 0: lanes map to contiguous channels for coalescing
  const int cf = tid & 7;           // channel within tile
  const int rf = tid >> 3;          // row within tile (0..31)
  const float* zg =
      z + ((size_t)(rb * RBLK + rf) * PADLEN + (size_t)(KTAP + s0)) * CDIM + c0 + cf;
  {
    const float* g = zg;
    float* l = &zbuf[0][cf * CSTR + rf * TPAD];
    #pragma unroll 7
    for (int j = 0; j < WIN; ++j) { async_load_b32(g, l); g += CDIM; ++l; }
  }
  wait_async0();
  __syncthreads();

  // ---- compute mapping: lane -> r (coalesced stores), wave -> channel
  const int r = tid & 31;
  const int c = tid >> 5;

  float2 acc[SB];
  #pragma unroll
  for (int i = 0; i < SB; ++i) acc[i] = make_float2(0.0f, 0.0f);

  for (int ck = 0; ck < NCHUNK; ++ck) {
    // prefetch next chunk into the other LDS buffer (overlaps with compute)
    if (ck + 1 < NCHUNK) {
      const float* g = zg + (size_t)(ck + 1) * KC * CDIM;
      float* l = &zbuf[(ck + 1) & 1][cf * CSTR + rf * TPAD];
      #pragma unroll 7
      for (int j = 0; j < WIN; ++j) { async_load_b32(g, l); g += CDIM; ++l; }
    }

    // pull this thread's 63-timestep window into registers (ds_load_b128)
    const float* zl = &zbuf[ck & 1][c * CSTR + r * TPAD];
    float zw[64];
    #pragma unroll
    for (int i = 0; i < 16; ++i)
      *(float4*)&zw[4 * i] = *(const float4*)&zl[4 * i];

    const float* wk_l = &wlds[(0 * CB + c) * KPAD + ck * KC];
    const float* wq_l = &wlds[(1 * CB + c) * KPAD + ck * KC];

    #pragma unroll
    for (int g8 = 0; g8 < 4; ++g8) {
      float wkv[8], wqv[8];
      *(float4*)&wkv[0] = *(const float4*)&wk_l[g8 * 8 + 0];
      *(float4*)&wkv[4] = *(const float4*)&wk_l[g8 * 8 + 4];
      *(float4*)&wqv[0] = *(const float4*)&wq_l[g8 * 8 + 0];
      *(float4*)&wqv[4] = *(const float4*)&wq_l[g8 * 8 + 4];
      #pragma unroll
      for (int k2 = 0; k2 < 8; ++k2) {
        const int   kk = g8 * 8 + k2;
        const float a  = wkv[k2];
        const float b  = wqv[k2];
        #pragma unroll
        for (int sl = 0; sl < SB; ++sl) {
          const float zv = zw[sl + kk];
          acc[sl].x = __builtin_fmaf(zv, a, acc[sl].x);
          acc[sl].y = __builtin_fmaf(zv, b, acc[sl].y);
        }
      }
    }

    wait_async0();       // next chunk's DMA done
    __syncthreads();     // everyone finished reading current buffer
  }

  // ---- scattered-but-lane-coalesced stores (32 consecutive r = 128B)
  const int r_abs = rb * RBLK + r;
  const int c_abs = c0 + c;
  #pragma unroll
  for (int sl = 0; sl < SB; ++sl) {
    const int sg = s0 + sl;
    const size_t idx = (size_t)((sg & 3) * CDIM + c_abs) * 32768u
                     + (size_t)(sg >> 2) * 64u + (unsigned)r_abs;
    outA[idx] = acc[sl].x * 0.125f;   // /SCALE, SCALE = 8
    outB[idx] = acc[sl].y * 0.125f;
  }
}

// ---------------- host launcher ----------------
extern "C" void kernel_launch(void* const* d_in, const int* in_sizes, int n_in,
                              void* d_out, int out_size, void* d_ws, size_t ws_size,
                              hipStream_t stream) {
  (void)in_sizes; (void)n_in; (void)out_size; (void)d_ws; (void)ws_size;
  const float* z  = (const float*)d_in[0];
  const float* wq = (const float*)d_in[1];
  const float* wk = (const float*)d_in[2];
  float* outA = (float*)d_out;                                  // finish(conv(z, w_k))
  float* outB = outA + (size_t)SDIM * CDIM * RDIM;              // finish(conv(z, w_q))
  dim3 grid(64 * 64 * 2);   // c-tiles * s-tiles * r-tiles
  dim3 block(256);
  hipLaunchKernelGGL(ConvSPE_kernel, grid, block, 0, stream, z, wq, wk, outA, outB);
}